// EGCFv2Model_57526791962953
// MI455X (gfx1250) — compile-verified
//
#include <hip/hip_runtime.h>

// EGCFv2 gather + per-row dual dot product for gfx1250 (MI455X).
// Memory-bound (AI ~0.25 FLOP/B): all design effort goes into coalesced 256B
// row gathers; the K=64 reduction rides the matrix pipe via
// V_WMMA_F32_16X16X4_F32 (diag of 16x16 tile product), which is free at this
// arithmetic intensity.

typedef float v2f __attribute__((ext_vector_type(2)));
typedef float v8f __attribute__((ext_vector_type(8)));

#define WAVES_PER_BLOCK 4
#define TILE 16
#define EMBED_K 64
// 64 + 4 pad: b64 WMMA-operand reads hit all 64 LDS banks exactly once.
#define LDS_STRIDE 68

// Stage one 16x64 tile of gathered rows into LDS, fully coalesced:
// lanes 0-15 cover row (2*it), lanes 16-31 cover row (2*it+1); each lane moves
// one float4 -> each global_load_b128 instruction fetches 2 whole rows (512B, 256B
// contiguous per row).
__device__ __forceinline__ void stage_tile(const float* __restrict__ T,
                                           int rowIdxReg,  // lane m holds row-index of tile row m
                                           float (*__restrict__ dst)[LDS_STRIDE],
                                           int m, int half)
{
#pragma unroll
    for (int it = 0; it < 8; ++it) {
        const int row = 2 * it + half;
        const int r = __shfl(rowIdxReg, row, 32);
        const float4 v = *(const float4*)(T + (size_t)r * EMBED_K + m * 4);
        *(float4*)(&dst[row][m * 4]) = v;
    }
}

__global__ __launch_bounds__(WAVES_PER_BLOCK * 32)
void egcf_v2_wmma_kernel(const float* __restrict__ Gu,
                         const float* __restrict__ Gi,
                         const float* __restrict__ Gut,
                         const float* __restrict__ Git,
                         const int* __restrict__ uidx,
                         const int* __restrict__ pidx,
                         float* __restrict__ out,
                         int B)
{
    __shared__ float sU[WAVES_PER_BLOCK][TILE][LDS_STRIDE];
    __shared__ float sV[WAVES_PER_BLOCK][TILE][LDS_STRIDE];

    const int lane = threadIdx.x & 31;
    const int w    = threadIdx.x >> 5;
    const int half = lane >> 4;   // 0: lanes 0-15, 1: lanes 16-31
    const int m    = lane & 15;

    const int tileBase = ((int)blockIdx.x * WAVES_PER_BLOCK + w) * TILE;
    if (tileBase >= B) return;    // wave-uniform: EXEC stays all-ones for WMMA

    // Lane m (both halves) holds the gather indices for tile row m (clamped
    // for the tail tile so no lane diverges on loads).
    const int il = min(tileBase + m, B - 1);
    const int iu = uidx[il];
    const int ip = pidx[il];

    v8f acc = {};   // 16x16 f32 accumulator: D = U*V^T + Ut*Vt^T

    // ---------------- pair 1: Gu[user] x Gi[pos] ----------------
    stage_tile(Gu, iu, sU[w], m, half);
    stage_tile(Gi, ip, sV[w], m, half);

    // A 16x4 f32 operand image: v0 = {K=0 | K=2}, v1 = {K=1 | K=3} per lane
    // halves; i.e. lane l reads tile[l&15][4c + 2*(l>>4) + {0,1}] -> one b64.
    // B (4x16) uses the mirrored K-striping, so both operands read with the
    // identical pattern and the WMMA computes U_tile * V_tile^T.
#pragma unroll
    for (int c = 0; c < 16; ++c) {
        const v2f a = *(const v2f*)(&sU[w][m][4 * c + 2 * half]);
        const v2f b = *(const v2f*)(&sV[w][m][4 * c + 2 * half]);
        acc = __builtin_amdgcn_wmma_f32_16x16x4_f32(
            /*neg_a=*/false, a, /*neg_b=*/false, b,
            /*c_mod=*/(short)0, acc, /*reuse_a=*/false, /*reuse_b=*/false);
    }

    // ---------------- pair 2: Gut[user] x Git[pos] ----------------
    // Same-wave DS ops are processed in order, so restaging over the buffers
    // after the pair-1 ds_loads is safe without a workgroup barrier.
    stage_tile(Gut, iu, sU[w], m, half);
    stage_tile(Git, ip, sV[w], m, half);

#pragma unroll
    for (int c = 0; c < 16; ++c) {
        const v2f a = *(const v2f*)(&sU[w][m][4 * c + 2 * half]);
        const v2f b = *(const v2f*)(&sV[w][m][4 * c + 2 * half]);
        acc = __builtin_amdgcn_wmma_f32_16x16x4_f32(
            false, a, false, b, (short)0, acc, false, false);
    }

    // Diagonal extraction. C layout: acc[j] holds (M=j, N=lane) on lanes 0-15
    // and (M=j+8, N=lane-16) on lanes 16-31. So D[m,m] lives at:
    //   m in 0..7  -> lane m,      acc[m]
    //   m in 8..15 -> lane m+16,   acc[m-8]
    const int j = lane & 7;
    float dv = acc[0];
    dv = (j == 1) ? acc[1] : dv;
    dv = (j == 2) ? acc[2] : dv;
    dv = (j == 3) ? acc[3] : dv;
    dv = (j == 4) ? acc[4] : dv;
    dv = (j == 5) ? acc[5] : dv;
    dv = (j == 6) ? acc[6] : dv;
    dv = (j == 7) ? acc[7] : dv;

    const int src = (m < 8) ? m : (m + 16);
    const float result = __shfl(dv, src, 32);

    if (lane < 16) {
        const int o = tileBase + lane;
        if (o < B) out[o] = result;
    }
}

extern "C" void kernel_launch(void* const* d_in, const int* in_sizes, int n_in,
                              void* d_out, int out_size, void* d_ws, size_t ws_size,
                              hipStream_t stream) {
    const float* Gu  = (const float*)d_in[0];
    const float* Gi  = (const float*)d_in[1];
    const float* Gut = (const float*)d_in[2];
    const float* Git = (const float*)d_in[3];
    const int*   ui  = (const int*)d_in[4];
    const int*   pi  = (const int*)d_in[5];
    float* out = (float*)d_out;

    const int B = in_sizes[4];  // user_idx element count == BATCH
    const int elemsPerBlock = WAVES_PER_BLOCK * TILE;
    const int grid = (B + elemsPerBlock - 1) / elemsPerBlock;

    hipLaunchKernelGGL(egcf_v2_wmma_kernel, dim3(grid), dim3(WAVES_PER_BLOCK * 32),
                       0, stream, Gu, Gi, Gut, Git, ui, pi, out, B);
}